// DAFRouter_32495722561931
// MI455X (gfx1250) — compile-verified
//
#include <hip/hip_runtime.h>
#include <hip/hip_bf16.h>
#include <math.h>

// DAF router: logits = [h | MLP(metadata)] @ wg + bg ; top-8 ; masked softmax.
// N=16384, D=4096, E=64, M_IN=2, M_H=16, M_OUT=8, k=8
//
// Memory-bound (268MB of h @ 23.3 TB/s ~= 11.5us floor), so we keep full f32
// precision and use V_WMMA_F32_16X16X4_F32 for the GEMM.

#define NROWS  16384
#define DDIM   4096
#define EEXP   64
#define KTOP   8

typedef float v2f __attribute__((ext_vector_type(2)));
typedef float v8f __attribute__((ext_vector_type(8)));

// ---------------------------------------------------------------------------
// Kernel 1: transpose wg[0:4096, 0:64] -> wgT[64][4096] (in d_ws) so the WMMA
// B operand (lane = N column, K-pair in v0/v1) is a single contiguous 8-byte
// load. Also copies mu to the output tail.
// ---------------------------------------------------------------------------
__global__ __launch_bounds__(256) void daf_transpose_wg(
    const float* __restrict__ wg, float* __restrict__ wgT,
    const float* __restrict__ mu, float* __restrict__ out)
{
    int idx = blockIdx.x * 256 + threadIdx.x;
    if (idx < DDIM * EEXP) {
        int n = idx >> 12;          // idx / 4096 -> column of wg
        int k = idx & (DDIM - 1);   // idx % 4096 -> row of wg
        wgT[idx] = wg[k * EEXP + n];
    }
    if (idx < EEXP) {
        out[(size_t)NROWS * EEXP + (size_t)NROWS * KTOP + idx] = mu[idx];
    }
}

// ---------------------------------------------------------------------------
// Kernel 2: main fused GEMM + epilogue.
// Block = 256 threads (8 waves). Block owns 64 rows = 4 strips of 16 rows.
// Wave w: strip = w&3, khalf = w>>2 (K split in two for 2x wave parallelism).
// ---------------------------------------------------------------------------
__global__ __launch_bounds__(256) void daf_router_main(
    const float* __restrict__ h,
    const float* __restrict__ metadata,
    const float* __restrict__ wg,     // full [4104,64] (for m_emb rows)
    const float* __restrict__ bg,
    const float* __restrict__ w1, const float* __restrict__ b1,
    const float* __restrict__ w2, const float* __restrict__ b2,
    const float* __restrict__ wgT,    // [64][4096] transposed
    float* __restrict__ out)
{
    __shared__ float ldsA[64 * 65];   // partial C, khalf 0 (stride 65: no bank conflicts)
    __shared__ float ldsB[64 * 65];   // partial C, khalf 1

    const int tid   = threadIdx.x;
    const int wave  = tid >> 5;
    const int lane  = tid & 31;
    const int strip = wave & 3;
    const int khalf = wave >> 2;
    const int m     = lane & 15;      // row-within-strip (A) / column-within-tile (B,C)
    const int grp   = lane >> 4;      // lane group: selects K pair / C row half

    const int rowBase = blockIdx.x * 64 + strip * 16;

    v8f acc0 = {0,0,0,0,0,0,0,0};
    v8f acc1 = {0,0,0,0,0,0,0,0};
    v8f acc2 = {0,0,0,0,0,0,0,0};
    v8f acc3 = {0,0,0,0,0,0,0,0};

    // A layout (16x4 f32): lanes 0-15 hold M=lane, v0=K+0,v1=K+1; lanes 16-31 v0=K+2,v1=K+3.
    const float* hrow = h + (size_t)(rowBase + m) * DDIM + 2 * grp;
    // B layout (4x16 f32): v0 = row K (+2 for grp1) across lanes(N), v1 = row K+1 (+2).
    const float* bt0 = wgT + (size_t)( 0 + m) * DDIM + 2 * grp;
    const float* bt1 = wgT + (size_t)(16 + m) * DDIM + 2 * grp;
    const float* bt2 = wgT + (size_t)(32 + m) * DDIM + 2 * grp;
    const float* bt3 = wgT + (size_t)(48 + m) * DDIM + 2 * grp;

    const int k0 = khalf * (DDIM / 2);

    for (int kb = 0; kb < DDIM / 2; kb += 16) {   // 4 WMMA K-steps per group
        // locality 3 -> WGP-scope prefetch: warm the near cache for the h stream
        __builtin_prefetch(hrow + k0 + kb + 256, 0, 3);
#pragma unroll
        for (int u = 0; u < 16; u += 4) {
            const int k = k0 + kb + u;
            v2f a  = *reinterpret_cast<const v2f*>(hrow + k);
            v2f b0 = *reinterpret_cast<const v2f*>(bt0 + k);
            v2f b1v= *reinterpret_cast<const v2f*>(bt1 + k);
            v2f b2v= *reinterpret_cast<const v2f*>(bt2 + k);
            v2f b3 = *reinterpret_cast<const v2f*>(bt3 + k);
            acc0 = __builtin_amdgcn_wmma_f32_16x16x4_f32(false, a, false, b0, (short)0, acc0, false, false);
            acc1 = __builtin_amdgcn_wmma_f32_16x16x4_f32(false, a, false, b1v,(short)0, acc1, false, false);
            acc2 = __builtin_amdgcn_wmma_f32_16x16x4_f32(false, a, false, b2v,(short)0, acc2, false, false);
            acc3 = __builtin_amdgcn_wmma_f32_16x16x4_f32(false, a, false, b3, (short)0, acc3, false, false);
        }
    }

    // C layout: element j of v8f = row (8*grp + j), column = c*16 + m.
    float* L = khalf ? ldsB : ldsA;
    {
        const int rl = strip * 16 + 8 * grp;
#pragma unroll
        for (int j = 0; j < 8; ++j) {
            L[(rl + j) * 65 +  0 + m] = acc0[j];
            L[(rl + j) * 65 + 16 + m] = acc1[j];
            L[(rl + j) * 65 + 32 + m] = acc2[j];
            L[(rl + j) * 65 + 48 + m] = acc3[j];
        }
    }
    __syncthreads();

    // ---------------- per-row epilogue: threads 0..63 own one row each -----
    if (tid < 64) {
        const int r = blockIdx.x * 64 + tid;

        // metadata MLP: 2 -> 16 (exact GELU) -> 8
        const float md0 = metadata[(size_t)r * 2 + 0];
        const float md1 = metadata[(size_t)r * 2 + 1];
        float memb[8];
#pragma unroll
        for (int q = 0; q < 8; ++q) memb[q] = b2[q];
#pragma unroll
        for (int j = 0; j < 16; ++j) {
            float x = md0 * w1[j] + md1 * w1[16 + j] + b1[j];
            float g = 0.5f * x * (1.0f + erff(x * 0.70710678118654752f));
#pragma unroll
            for (int q = 0; q < 8; ++q) memb[q] += g * w2[j * 8 + q];
        }

        // finalize logits in ldsA: partialA + partialB + bg + m_emb @ wg[4096: , :]
        for (int i = 0; i < EEXP; ++i) {
            float v = ldsA[tid * 65 + i] + ldsB[tid * 65 + i] + bg[i];
#pragma unroll
            for (int q = 0; q < 8; ++q) v += memb[q] * wg[(size_t)(DDIM + q) * EEXP + i];
            ldsA[tid * 65 + i] = v;
        }

        // top-8 (strict '>' => lowest index wins ties, matching lax.top_k)
        float topv[KTOP];
        int   topi[KTOP];
#pragma unroll
        for (int p = 0; p < KTOP; ++p) {
            float best = -INFINITY; int bi = 0;
            for (int i = 0; i < EEXP; ++i) {
                float v = ldsA[tid * 65 + i];
                if (v > best) { best = v; bi = i; }
            }
            topv[p] = best; topi[p] = bi;
            ldsA[tid * 65 + bi] = -INFINITY;
        }

        // softmax over the selected 8 (all others are exactly 0)
        const float mx = topv[0];
        float e[KTOP], s = 0.0f;
#pragma unroll
        for (int p = 0; p < KTOP; ++p) { e[p] = expf(topv[p] - mx); s += e[p]; }
        const float inv = 1.0f / s;

        // rewrite row in LDS with gating weights
        for (int i = 0; i < EEXP; ++i) ldsA[tid * 65 + i] = 0.0f;
#pragma unroll
        for (int p = 0; p < KTOP; ++p) ldsA[tid * 65 + topi[p]] = e[p] * inv;

        // indices output (as float, concatenated after gating weights)
        float* oi = out + (size_t)NROWS * EEXP;
#pragma unroll
        for (int p = 0; p < KTOP; ++p) oi[(size_t)r * KTOP + p] = (float)topi[p];
    }
    __syncthreads();

    // coalesced write of the 64x64 gating tile
    for (int e2 = tid; e2 < 64 * EEXP; e2 += 256) {
        int rl = e2 >> 6, c = e2 & 63;
        out[(size_t)(blockIdx.x * 64 + rl) * EEXP + c] = ldsA[rl * 65 + c];
    }
}

// ---------------------------------------------------------------------------
// Host launcher. Input order: h, metadata, k, w1, b1, w2, b2, wg, bg, mu.
// d_ws needs 4096*64*4 = 1 MiB for the transposed gate weights.
// ---------------------------------------------------------------------------
extern "C" void kernel_launch(void* const* d_in, const int* in_sizes, int n_in,
                              void* d_out, int out_size, void* d_ws, size_t ws_size,
                              hipStream_t stream)
{
    const float* h        = (const float*)d_in[0];
    const float* metadata = (const float*)d_in[1];
    /* d_in[2] is k == 8, baked into KTOP */
    const float* w1 = (const float*)d_in[3];
    const float* b1 = (const float*)d_in[4];
    const float* w2 = (const float*)d_in[5];
    const float* b2 = (const float*)d_in[6];
    const float* wg = (const float*)d_in[7];
    const float* bg = (const float*)d_in[8];
    const float* mu = (const float*)d_in[9];
    float* out = (float*)d_out;
    float* wgT = (float*)d_ws;    // 1 MiB

    daf_transpose_wg<<<(DDIM * EEXP + 255) / 256, 256, 0, stream>>>(wg, wgT, mu, out);
    daf_router_main<<<NROWS / 64, 256, 0, stream>>>(h, metadata, wg, bg, w1, b1, w2, b2, wgT, out);
}